// UDESolver_15951508537806
// MI455X (gfx1250) — compile-verified
//
#include <hip/hip_runtime.h>

typedef __attribute__((ext_vector_type(2))) float v2f;
typedef __attribute__((ext_vector_type(8))) float v8f;

#define DIM 32
#define HID 128
#define NTV 64
#define NIN 8

#define A21f 0.161f
#define A31f (-0.008480655492356989f)
#define A32f 0.335480655492357f
#define A41f 2.8971530571054935f
#define A42f (-6.359448489975075f)
#define A43f 4.3622954328695815f
#define A51f 5.325864828439257f
#define A52f (-11.748883564062828f)
#define A53f 7.4955393428898365f
#define A54f (-0.09249506636175525f)
#define A61f 5.86145544294642f
#define A62f (-12.92096931784711f)
#define A63f 8.159367898576159f
#define A64f (-0.071584973281401f)
#define A65f (-0.028269050394068383f)
#define B1f 0.09646076681806523f
#define B2f 0.01f
#define B3f 0.4798896504144996f
#define B4f 1.379008574103742f
#define B5f (-3.290069515436081f)
#define B6f 2.324710524099774f
#define E1f (-0.001780011052225777f)
#define E2f (-0.0008164344596567469f)
#define E3f 0.007880878010261995f
#define E4f (-0.1447110071732629f)
#define E5f 0.5823571654525552f
#define E6f (-0.45808210592918697f)
#define E7f 0.015151515151515152f

__device__ const float AC[6][6] = {
  {A21f, 0.f,  0.f,  0.f,  0.f,  0.f},
  {A31f, A32f, 0.f,  0.f,  0.f,  0.f},
  {A41f, A42f, A43f, 0.f,  0.f,  0.f},
  {A51f, A52f, A53f, A54f, 0.f,  0.f},
  {A61f, A62f, A63f, A64f, A65f, 0.f},
  {B1f,  B2f,  B3f,  B4f,  B5f,  B6f},
};

static __device__ __forceinline__ v8f wmma4(v2f a, v2f b, v8f c) {
  // D = A(16x4,f32) * B(4x16,f32) + C(16x16,f32)  -> v_wmma_f32_16x16x4_f32
  return __builtin_amdgcn_wmma_f32_16x16x4_f32(false, a, false, b, (short)0, c,
                                               false, false);
}

static __device__ __forceinline__ float tanh_fast(float x) {
#if __has_builtin(__builtin_amdgcn_tanhf)
  return __builtin_amdgcn_tanhf(x);
#elif __has_builtin(__builtin_amdgcn_tanh_f32)
  return __builtin_amdgcn_tanh_f32(x);
#else
  return tanhf(x);
#endif
}

// vf(z) = -0.5*z + tanh(z@W1 + b1)@W2 + b2 for a 16-row tile, one wave32.
// z, kout, hs are LDS. Weight B-fragments live in registers (w1f, w2f).
static __device__ __forceinline__ void vf_tile(
    const float* z, float* kout, float* hs,
    const v2f* w1f, const v2f* w2f,
    const float* b1v, const float* b2v,
    int r, int half)
{
  // A fragments of z: lane r holds row r; VGPR0/1 = K = 4k+2*half (+1)
  v2f az[8];
#pragma unroll
  for (int k = 0; k < 8; ++k)
    az[k] = *(const v2f*)(z + r * DIM + 4 * k + 2 * half);

  // GEMM1: h(16x128) = z @ W1, C initialized with b1 broadcast
  v8f h[8];
#pragma unroll
  for (int n = 0; n < 8; ++n) {
    const float bb = b1v[n];
    v8f hn = {bb, bb, bb, bb, bb, bb, bb, bb};
    h[n] = hn;
  }
#pragma unroll
  for (int k = 0; k < 8; ++k)
#pragma unroll
    for (int n = 0; n < 8; ++n)
      h[n] = wmma4(az[k], w1f[k * 8 + n], h[n]);

  // tanh, then transpose C-layout -> row-major via LDS
#pragma unroll
  for (int n = 0; n < 8; ++n)
#pragma unroll
    for (int j = 0; j < 8; ++j)
      hs[(8 * half + j) * HID + 16 * n + r] = tanh_fast(h[n][j]);

  // GEMM2: out(16x32) = h @ W2, C initialized with b2 broadcast
  v8f o0, o1;
  { const float bb = b2v[0]; v8f t = {bb, bb, bb, bb, bb, bb, bb, bb}; o0 = t; }
  { const float bb = b2v[1]; v8f t = {bb, bb, bb, bb, bb, bb, bb, bb}; o1 = t; }
#pragma unroll
  for (int k = 0; k < 32; ++k) {
    const v2f ah = *(const v2f*)(hs + r * HID + 4 * k + 2 * half);
    o0 = wmma4(ah, w2f[2 * k + 0], o0);
    o1 = wmma4(ah, w2f[2 * k + 1], o1);
  }

  // k = out - 0.5*z (C layout: lane owns column r|r+16, rows 8*half+j)
#pragma unroll
  for (int j = 0; j < 8; ++j) {
    const int m = 8 * half + j;
    kout[m * DIM +      r] = o0[j] - 0.5f * z[m * DIM +      r];
    kout[m * DIM + 16 + r] = o1[j] - 0.5f * z[m * DIM + 16 + r];
  }
}

__global__ void __launch_bounds__(32, 1)
ude_rk45_kernel(const float* __restrict__ y0, const float* __restrict__ ts,
                const float* __restrict__ W1, const float* __restrict__ b1,
                const float* __restrict__ W2, const float* __restrict__ b2,
                float* __restrict__ out)
{
  __shared__ __align__(16) float ys[16 * DIM];
  __shared__ __align__(16) float yn[16 * DIM];
  __shared__ __align__(16) float zs[16 * DIM];
  __shared__ __align__(16) float ksAll[7 * 16 * DIM];  // k1..k7
  __shared__ __align__(16) float hs[16 * HID];
  __shared__ float tA[16], dtA[16], dtcA[16], accA[16];

  const int lane = (int)threadIdx.x;
  const int r    = lane & 15;
  const int half = lane >> 4;
  const int b0   = (int)blockIdx.x * 16;

  // Pre-swizzled weight B-fragments, resident in VGPRs for the whole solve.
  v2f w1f[64];  // [kstep 0..7][ntile 0..7]
#pragma unroll
  for (int k = 0; k < 8; ++k)
#pragma unroll
    for (int n = 0; n < 8; ++n) {
      const int row = 4 * k + 2 * half;
      v2f w = { W1[row * HID + 16 * n + r], W1[(row + 1) * HID + 16 * n + r] };
      w1f[k * 8 + n] = w;
    }
  v2f w2f[64];  // [kstep 0..31][ntile 0..1]
#pragma unroll
  for (int k = 0; k < 32; ++k)
#pragma unroll
    for (int t = 0; t < 2; ++t) {
      const int row = 4 * k + 2 * half;
      v2f w = { W2[row * DIM + 16 * t + r], W2[(row + 1) * DIM + 16 * t + r] };
      w2f[2 * k + t] = w;
    }
  float b1v[8];
#pragma unroll
  for (int n = 0; n < 8; ++n) b1v[n] = b1[16 * n + r];
  float b2v[2] = { b2[r], b2[16 + r] };

  // Load y0 tile; emit trajectory sample 0.
#pragma unroll
  for (int j = 0; j < 16; ++j) {
    const int c = 16 * half + j;
    const float v = y0[(b0 + r) * DIM + c];
    ys[r * DIM + c] = v;
    out[((size_t)(b0 + r) * NTV + 0) * DIM + c] = v;
  }
  if (lane < 16) { tA[r] = ts[0]; dtA[r] = 0.05f; }
  __syncthreads();

  // k1 = vf(y0)
  vf_tile(ys, &ksAll[0], hs, w1f, w2f, b1v, b2v, r, half);

#pragma unroll 1
  for (int it = 1; it < NTV; ++it) {
    const float t_b = ts[it];
#pragma unroll 1
    for (int inner = 0; inner < NIN; ++inner) {
      float t_r = 0.f, dt_r = 0.f, dtc_r = 0.f;
      bool done_r = false;
      if (lane < 16) {
        t_r = tA[r];
        dt_r = dtA[r];
        done_r = (t_r >= t_b - 1e-9f);
        dtc_r = fmaxf(fminf(dt_r, t_b - t_r), 1e-12f);
        dtcA[r] = dtc_r;
      }
      const float d = dtcA[r];  // same-wave LDS store->load is in-order

      // Stages: s=0..4 -> k2..k6 (input zs); s=5 -> y_new (yn) and k7.
#pragma unroll 1
      for (int s = 0; s < 6; ++s) {
        float* zt = (s == 5) ? yn : zs;
#pragma unroll
        for (int j = 0; j < 16; ++j) {
          const int i = r * DIM + 16 * half + j;
          float a = 0.f;
          for (int p = 0; p <= s; ++p) a += AC[s][p] * ksAll[p * 512 + i];
          zt[i] = ys[i] + d * a;
        }
        vf_tile(zt, &ksAll[(s + 1) * 512], hs, w1f, w2f, b1v, b2v, r, half);
      }

      // err -> zs
#pragma unroll
      for (int j = 0; j < 16; ++j) {
        const int i = r * DIM + 16 * half + j;
        const float e =
            E1f * ksAll[0 * 512 + i] + E2f * ksAll[1 * 512 + i] +
            E3f * ksAll[2 * 512 + i] + E4f * ksAll[3 * 512 + i] +
            E5f * ksAll[4 * 512 + i] + E6f * ksAll[5 * 512 + i] +
            E7f * ksAll[6 * 512 + i];
        zs[i] = d * e;
      }

      // Per-row error norm + controller (lanes 0..15, one row each)
      if (lane < 16) {
        float ssum = 0.f;
        for (int c = 0; c < DIM; ++c) {
          const float e = zs[r * DIM + c];
          const float tol = 1e-6f + 1e-3f * fmaxf(fabsf(ys[r * DIM + c]),
                                                  fabsf(yn[r * DIM + c]));
          const float q = e / tol;
          ssum += q * q;
        }
        const float en = sqrtf(ssum * (1.0f / 32.0f));
        const bool accept = (en <= 1.0f) && !done_r;
        const float enc = fmaxf(en, 1e-10f);
        float factor = 0.9f * expf(-0.2f * logf(enc));  // SAFETY * en^-0.2
        factor = fminf(fmaxf(factor, 0.1f), 5.0f);
        dtA[r] = done_r ? dt_r : dtc_r * factor;
        tA[r] = accept ? (t_r + dtc_r) : t_r;
        accA[r] = accept ? 1.f : 0.f;
      }
      const float acc = accA[r];
#pragma unroll
      for (int j = 0; j < 16; ++j) {
        const int i = r * DIM + 16 * half + j;
        const float yv = (acc != 0.f) ? yn[i] : ys[i];
        const float kv = (acc != 0.f) ? ksAll[6 * 512 + i] : ksAll[i];
        ys[i] = yv;
        ksAll[i] = kv;  // k1 <- k7 on accept (FSAL)
      }
    }

    // Emit trajectory sample `it`
#pragma unroll
    for (int j = 0; j < 16; ++j) {
      const int c = 16 * half + j;
      out[((size_t)(b0 + r) * NTV + it) * DIM + c] = ys[r * DIM + c];
    }
  }
}

extern "C" void kernel_launch(void* const* d_in, const int* in_sizes, int n_in,
                              void* d_out, int out_size, void* d_ws, size_t ws_size,
                              hipStream_t stream) {
  (void)n_in; (void)out_size; (void)d_ws; (void)ws_size;
  const float* y0 = (const float*)d_in[0];
  const float* ts = (const float*)d_in[1];
  const float* W1 = (const float*)d_in[2];
  const float* b1 = (const float*)d_in[3];
  const float* W2 = (const float*)d_in[4];
  const float* b2 = (const float*)d_in[5];
  float* out = (float*)d_out;
  const int B = in_sizes[0] / DIM;      // 512
  const int nblocks = B / 16;           // 32 waves total, whole solve on-chip
  ude_rk45_kernel<<<nblocks, 32, 0, stream>>>(y0, ts, W1, b1, W2, b2, out);
}